// PointPatchEmbed_78323023610180
// MI455X (gfx1250) — compile-verified
//
#include <hip/hip_runtime.h>
#include <hip/hip_bf16.h>

// Problem constants (match reference)
#define BB 32
#define NP 32768
#define GG 64
#define KK 32
#define DD 384
#define EPSV 1e-5f

typedef __attribute__((ext_vector_type(2))) float v2f;
typedef __attribute__((ext_vector_type(8))) float v8f;

// ---------------------------------------------------------------------------
// Warp (wave32) argmax/argmin with lowest-index tie-break (matches jnp.argmax
// / lax.top_k ordering). Result valid in lane 0.
// ---------------------------------------------------------------------------
__device__ __forceinline__ void warp_argmax(float& v, int& i) {
#pragma unroll
  for (int off = 16; off > 0; off >>= 1) {
    float ov = __shfl_down(v, off, 32);
    int oi = __shfl_down(i, off, 32);
    if (ov > v || (ov == v && oi < i)) { v = ov; i = oi; }
  }
}

__device__ __forceinline__ void warp_argmin(float& v, int& i) {
#pragma unroll
  for (int off = 16; off > 0; off >>= 1) {
    float ov = __shfl_down(v, off, 32);
    int oi = __shfl_down(i, off, 32);
    if (ov < v || (ov == v && oi < i)) { v = ov; i = oi; }
  }
}

// ---------------------------------------------------------------------------
// Kernel 1: Farthest point sampling. One block per batch. Running min-distance
// array (N floats = 128KB) kept in LDS (CDNA5: 320KB/WGP). Writes centers
// directly into the output buffer (first B*G*3 floats).
// ---------------------------------------------------------------------------
__global__ __launch_bounds__(1024) void fps_kernel(const float* __restrict__ xyz,
                                                   float* __restrict__ out_center) {
  extern __shared__ char smem_raw[];
  float* dist = (float*)smem_raw;          // NP floats
  float* rv = dist + NP;                   // 32 floats (per-wave partials)
  int* ri = (int*)(rv + 32);               // 32 ints
  int* cur = ri + 32;                      // 1 int

  const int b = blockIdx.x;
  const int tid = threadIdx.x;
  const int T = blockDim.x;
  const int lane = tid & 31;
  const int wav = tid >> 5;
  const int nw = T >> 5;
  const float* X = xyz + (size_t)b * NP * 3;

  for (int p = tid; p < NP; p += T) dist[p] = 1e10f;
  __syncthreads();

  int far = 0;
  for (int it = 0; it < GG; ++it) {
    // centroid = xyz[b][far] (uniform load, scalar path)
    float cx = X[far * 3 + 0];
    float cy = X[far * 3 + 1];
    float cz = X[far * 3 + 2];
    if (tid == 0) {
      float* oc = out_center + ((size_t)b * GG + it) * 3;
      oc[0] = cx; oc[1] = cy; oc[2] = cz;
    }
    float bv = -1.0f;
    int bi = 0x7fffffff;
    for (int p = tid; p < NP; p += T) {
      float dx = __fsub_rn(X[p * 3 + 0], cx);
      float dy = __fsub_rn(X[p * 3 + 1], cy);
      float dz = __fsub_rn(X[p * 3 + 2], cz);
      float d = __fadd_rn(__fadd_rn(__fmul_rn(dx, dx), __fmul_rn(dy, dy)),
                          __fmul_rn(dz, dz));
      float nd = fminf(dist[p], d);
      dist[p] = nd;  // each index touched by exactly one thread -> no race
      if (nd > bv) { bv = nd; bi = p; }
    }
    warp_argmax(bv, bi);
    if (lane == 0) { rv[wav] = bv; ri[wav] = bi; }
    __syncthreads();
    if (wav == 0) {
      float v = (lane < nw) ? rv[lane] : -1.0f;
      int i = (lane < nw) ? ri[lane] : 0x7fffffff;
      warp_argmax(v, i);
      if (lane == 0) *cur = i;
    }
    __syncthreads();
    far = *cur;
    __syncthreads();
  }
}

// ---------------------------------------------------------------------------
// Kernel 2: exact KNN (K=32 smallest d2, lowest-index ties) per (b,g).
// All N distances staged in LDS, then 32 argmin+mask passes.
// d2 computed exactly like the reference: (cn2 + xn2) - 2*dot (no fma).
// ---------------------------------------------------------------------------
__global__ __launch_bounds__(256) void knn_kernel(const float* __restrict__ xyz,
                                                  const float* __restrict__ centers,
                                                  int* __restrict__ knn_idx) {
  extern __shared__ char smem_raw[];
  float* d2 = (float*)smem_raw;  // NP floats
  float* rv = d2 + NP;           // 32
  int* ri = (int*)(rv + 32);     // 32
  int* cur = ri + 32;            // 1

  const int g = blockIdx.x;
  const int b = blockIdx.y;
  const int tid = threadIdx.x;
  const int T = blockDim.x;
  const int lane = tid & 31;
  const int wav = tid >> 5;
  const int nw = T >> 5;

  const float* X = xyz + (size_t)b * NP * 3;
  const float* C = centers + ((size_t)b * GG + g) * 3;
  const float cx = C[0], cy = C[1], cz = C[2];
  const float cn2 = __fadd_rn(__fadd_rn(__fmul_rn(cx, cx), __fmul_rn(cy, cy)),
                              __fmul_rn(cz, cz));

  for (int p = tid; p < NP; p += T) {
    float x = X[p * 3 + 0], y = X[p * 3 + 1], z = X[p * 3 + 2];
    float xn2 = __fadd_rn(__fadd_rn(__fmul_rn(x, x), __fmul_rn(y, y)), __fmul_rn(z, z));
    float dot = __fadd_rn(__fadd_rn(__fmul_rn(cx, x), __fmul_rn(cy, y)), __fmul_rn(cz, z));
    d2[p] = __fsub_rn(__fadd_rn(cn2, xn2), __fmul_rn(2.0f, dot));
  }
  __syncthreads();

  for (int k = 0; k < KK; ++k) {
    float bv = 3.4e38f;
    int bi = 0x7fffffff;
    for (int p = tid; p < NP; p += T) {
      float d = d2[p];
      if (d < bv) { bv = d; bi = p; }
    }
    warp_argmin(bv, bi);
    if (lane == 0) { rv[wav] = bv; ri[wav] = bi; }
    __syncthreads();
    if (wav == 0) {
      float v = (lane < nw) ? rv[lane] : 3.4e38f;
      int i = (lane < nw) ? ri[lane] : 0x7fffffff;
      warp_argmin(v, i);
      if (lane == 0) *cur = i;
    }
    __syncthreads();
    int sel = *cur;
    if (tid == 0) {
      knn_idx[((size_t)b * GG + g) * KK + k] = sel;
      d2[sel] = 3.0e38f;  // exclude from further passes
    }
    __syncthreads();
  }
}

// ---------------------------------------------------------------------------
// Kernel 3: gather + 3-layer pointwise MLP via V_WMMA_F32_16X16X4_F32 + maxpool.
// One block (8 waves) per (b,g) group of K=32 points.
//   A (16x4 f32) per-lane: row = lane&15, A[v] = X[row][kb + 2*(lane>>4) + v]
//   B (4x16 f32) per-lane: col = lane&15, B[v] = W[col][kb + 2*(lane>>4) + v]
//   C/D (16x16 f32):       n  = lane&15, m = v + 8*(lane>>4)
// BN folded to scale/bias; ReLU fused; K-maxpool fused into layer-3 epilogue.
// ---------------------------------------------------------------------------
#define H0S 8
#define H1S 68   // 64 + 4 pad -> conflict-free A-operand LDS reads
#define H2S 132  // 128 + 4 pad

__device__ __forceinline__ v8f wmma_f32(v2f a, v2f b, v8f c) {
  return __builtin_amdgcn_wmma_f32_16x16x4_f32(false, a, false, b, (short)0, c,
                                               false, false);
}

__global__ __launch_bounds__(256) void embed_kernel(
    const float* __restrict__ xyz, const float* __restrict__ feat,
    const float* __restrict__ centers, const int* __restrict__ knn_idx,
    const float* __restrict__ w1, const float* __restrict__ g1,
    const float* __restrict__ b1, const float* __restrict__ m1,
    const float* __restrict__ v1, const float* __restrict__ w2,
    const float* __restrict__ g2, const float* __restrict__ b2,
    const float* __restrict__ m2, const float* __restrict__ v2,
    const float* __restrict__ w3, const float* __restrict__ g3,
    const float* __restrict__ b3, const float* __restrict__ m3,
    const float* __restrict__ v3, float* __restrict__ out_patch) {
  __shared__ float h0[32 * H0S];
  __shared__ float w1p[64 * 8];  // W1 zero-padded 6 -> 8 along C
  __shared__ float h1[32 * H1S];
  __shared__ float h2[32 * H2S];
  __shared__ float s1[64], t1[64], s2[128], t2[128], s3[384], t3[384];

  const int g = blockIdx.x;
  const int b = blockIdx.y;
  const size_t bg = (size_t)b * GG + g;
  const int tid = threadIdx.x;
  const int lane = tid & 31;
  const int wav = tid >> 5;
  const int lr = lane & 15;   // row (A) / col (B,C,D) within tile
  const int lh = lane >> 4;   // half-wave select

  // Fold BatchNorm: y = x*s + t  with s = g*rsqrt(v+eps), t = b - m*s
  for (int c = tid; c < 64; c += 256) {
    float s = g1[c] * rsqrtf(v1[c] + EPSV); s1[c] = s; t1[c] = b1[c] - m1[c] * s;
  }
  for (int c = tid; c < 128; c += 256) {
    float s = g2[c] * rsqrtf(v2[c] + EPSV); s2[c] = s; t2[c] = b2[c] - m2[c] * s;
  }
  for (int c = tid; c < 384; c += 256) {
    float s = g3[c] * rsqrtf(v3[c] + EPSV); s3[c] = s; t3[c] = b3[c] - m3[c] * s;
  }
  for (int i = tid; i < 64 * 8; i += 256) {
    int o = i >> 3, c = i & 7;
    w1p[i] = (c < 6) ? w1[o * 6 + c] : 0.0f;
  }
  // Gather K=32 neighbors: h0 = [gx - center, gf], padded to 8 channels
  const float* C = centers + bg * 3;
  if (tid < 32) {
    int idx = knn_idx[bg * KK + tid];
    const float* xp = xyz + ((size_t)b * NP + idx) * 3;
    const float* fp = feat + ((size_t)b * NP + idx) * 3;
    h0[tid * H0S + 0] = xp[0] - C[0];
    h0[tid * H0S + 1] = xp[1] - C[1];
    h0[tid * H0S + 2] = xp[2] - C[2];
    h0[tid * H0S + 3] = fp[0];
    h0[tid * H0S + 4] = fp[1];
    h0[tid * H0S + 5] = fp[2];
    h0[tid * H0S + 6] = 0.0f;
    h0[tid * H0S + 7] = 0.0f;
  }
  __syncthreads();

  // ---- Layer 1: (32x8) x (8x64) -> h1(32x64), 8 tile-jobs over 8 waves ----
  {
    const int nt = wav >> 1, mt = wav & 1;
    v8f acc = {0.f, 0.f, 0.f, 0.f, 0.f, 0.f, 0.f, 0.f};
#pragma unroll
    for (int kb = 0; kb < 8; kb += 4) {
      const int ko = kb + 2 * lh;
      v2f a, w;
      a[0] = h0[(mt * 16 + lr) * H0S + ko];
      a[1] = h0[(mt * 16 + lr) * H0S + ko + 1];
      w[0] = w1p[(nt * 16 + lr) * 8 + ko];
      w[1] = w1p[(nt * 16 + lr) * 8 + ko + 1];
      acc = wmma_f32(a, w, acc);
    }
    const float sc = s1[nt * 16 + lr], bi = t1[nt * 16 + lr];
#pragma unroll
    for (int v = 0; v < 8; ++v) {
      int row = mt * 16 + v + 8 * lh;
      h1[row * H1S + nt * 16 + lr] = fmaxf(acc[v] * sc + bi, 0.0f);
    }
  }
  __syncthreads();

  // ---- Layer 2: (32x64) x (64x128) -> h2(32x128), 16 tile-jobs ----
  for (int job = wav; job < 16; job += 8) {
    const int nt = job >> 1, mt = job & 1;
    v8f acc = {0.f, 0.f, 0.f, 0.f, 0.f, 0.f, 0.f, 0.f};
#pragma unroll 4
    for (int kb = 0; kb < 64; kb += 4) {
      const int ko = kb + 2 * lh;
      v2f a, w;
      a[0] = h1[(mt * 16 + lr) * H1S + ko];
      a[1] = h1[(mt * 16 + lr) * H1S + ko + 1];
      const float* wp = w2 + (nt * 16 + lr) * 64 + ko;
      w[0] = wp[0];
      w[1] = wp[1];
      acc = wmma_f32(a, w, acc);
    }
    const float sc = s2[nt * 16 + lr], bi = t2[nt * 16 + lr];
#pragma unroll
    for (int v = 0; v < 8; ++v) {
      int row = mt * 16 + v + 8 * lh;
      h2[row * H2S + nt * 16 + lr] = fmaxf(acc[v] * sc + bi, 0.0f);
    }
  }
  __syncthreads();

  // ---- Layer 3: (32x128) x (128x384) + fused max over K=32 rows ----
  for (int nt = wav; nt < 24; nt += 8) {
    v8f acc0 = {0.f, 0.f, 0.f, 0.f, 0.f, 0.f, 0.f, 0.f};
    v8f acc1 = {0.f, 0.f, 0.f, 0.f, 0.f, 0.f, 0.f, 0.f};
#pragma unroll 4
    for (int kb = 0; kb < 128; kb += 4) {
      const int ko = kb + 2 * lh;
      v2f a0, a1, w;
      a0[0] = h2[lr * H2S + ko];
      a0[1] = h2[lr * H2S + ko + 1];
      a1[0] = h2[(16 + lr) * H2S + ko];
      a1[1] = h2[(16 + lr) * H2S + ko + 1];
      const float* wp = w3 + (nt * 16 + lr) * 128 + ko;
      w[0] = wp[0];
      w[1] = wp[1];
      acc0 = wmma_f32(a0, w, acc0);
      acc1 = wmma_f32(a1, w, acc1);
    }
    const float sc = s3[nt * 16 + lr], bi = t3[nt * 16 + lr];
    float r = -3.4e38f;
#pragma unroll
    for (int v = 0; v < 8; ++v) {
      r = fmaxf(r, acc0[v] * sc + bi);
      r = fmaxf(r, acc1[v] * sc + bi);
    }
    r = fmaxf(r, __shfl_xor(r, 16, 32));  // combine the two row-halves
    if (lh == 0) out_patch[bg * DD + nt * 16 + lr] = r;
  }
}

// ---------------------------------------------------------------------------
// Launch
// ---------------------------------------------------------------------------
extern "C" void kernel_launch(void* const* d_in, const int* in_sizes, int n_in,
                              void* d_out, int out_size, void* d_ws, size_t ws_size,
                              hipStream_t stream) {
  (void)in_sizes; (void)n_in; (void)out_size; (void)ws_size;
  const float* xyz = (const float*)d_in[0];
  const float* feat = (const float*)d_in[1];
  const float* w1 = (const float*)d_in[2];
  const float* g1 = (const float*)d_in[3];
  const float* b1 = (const float*)d_in[4];
  const float* m1 = (const float*)d_in[5];
  const float* v1 = (const float*)d_in[6];
  const float* w2 = (const float*)d_in[7];
  const float* g2 = (const float*)d_in[8];
  const float* b2 = (const float*)d_in[9];
  const float* m2 = (const float*)d_in[10];
  const float* v2 = (const float*)d_in[11];
  const float* w3 = (const float*)d_in[12];
  const float* g3 = (const float*)d_in[13];
  const float* b3 = (const float*)d_in[14];
  const float* m3 = (const float*)d_in[15];
  const float* v3 = (const float*)d_in[16];

  float* out = (float*)d_out;
  float* out_center = out;             // (B,G,3)
  float* out_patch = out + BB * GG * 3;  // (B,G,D)
  int* knn = (int*)d_ws;               // (B,G,K) ints

  const size_t smem_scan = (size_t)(NP + 80) * sizeof(float);

  fps_kernel<<<dim3(BB), dim3(1024), smem_scan, stream>>>(xyz, out_center);
  knn_kernel<<<dim3(GG, BB), dim3(256), smem_scan, stream>>>(xyz, out_center, knn);
  embed_kernel<<<dim3(GG, BB), dim3(256), 0, stream>>>(
      xyz, feat, out_center, knn, w1, g1, b1, m1, v1, w2, g2, b2, m2, v2, w3,
      g3, b3, m3, v3, out_patch);
}